// interGATLayer_59528246722752
// MI455X (gfx1250) — compile-verified
//
#include <hip/hip_runtime.h>
#include <math.h>

// ---------------------------------------------------------------------------
// interGATLayer for MI455X (gfx1250). All GEMMs use V_WMMA_F32_16X16X4_F32;
// 64KB weight staging uses the Tensor Data Mover (tensor_load_to_lds).
// Shapes: N=20000 nodes, E=320000 edges, feature dims 128 everywhere.
// Everything is f32 (memory-bound pipeline; precision matches reference).
// ---------------------------------------------------------------------------

typedef __attribute__((ext_vector_type(2))) float v2f;
typedef __attribute__((ext_vector_type(8))) float v8f;
typedef __attribute__((ext_vector_type(4))) unsigned int u32x4;
typedef __attribute__((ext_vector_type(8))) int i32x8;
typedef __attribute__((ext_vector_type(4))) int i32x4;

#define N_NODES 20000
#define N_EDGES 320000
#define DIM 128

static __device__ __forceinline__ float lrelu(float x) { return x >= 0.f ? x : 0.1f * x; }

static __device__ __forceinline__ v8f wmma4(v2f a, v2f b, v8f c) {
  // D = A(16x4) * B(4x16) + C, f32. (neg_a, A, neg_b, B, c_mod, C, reuse_a, reuse_b)
  return __builtin_amdgcn_wmma_f32_16x16x4_f32(false, a, false, b, (short)0, c,
                                               false, false);
}

// --------- TDM: DMA a K x 128 f32 row-major tile global -> LDS -------------
// D# per CDNA5 ISA. 2-D tensor: groups 2/3 zero.
// Issued by ONE wave; completion via s_wait_tensorcnt, then workgroup barrier.
// This toolchain exposes the 6-arg builtin:
//   (u32x4 g0, i32x8 g1, i32x4 g2, i32x4 g3, i32x8 extra, i32 cpol)
#if __has_builtin(__builtin_amdgcn_tensor_load_to_lds)
#define HAVE_TDM 1
static __device__ __forceinline__ void tdm_load_Kx128(const float* g,
                                                      unsigned lds_off,
                                                      unsigned Krows) {
  const unsigned long long ga = (unsigned long long)(uintptr_t)g;
  u32x4 g0;
  g0[0] = 1u;                                               // count=1, user D#
  g0[1] = lds_off;                                          // lds_addr
  g0[2] = (unsigned)ga;                                     // global_addr[31:0]
  g0[3] = (unsigned)((ga >> 32) & 0x01FFFFFFu) | (2u << 30);// ga[56:32], type=2
  i32x8 g1;
  g1[0] = (int)(2u << 16);          // wg_mask=0, data_size=2 (4B), no pad/iter
  g1[1] = (int)(128u << 16);        // tensor_dim0=128 (bits 79:48)
  g1[2] = (int)(Krows << 16);       // tensor_dim0 hi=0 | tensor_dim1=K lo16
  g1[3] = (int)(128u << 16);        // tensor_dim1 hi=0 | tile_dim0=128
  g1[4] = (int)Krows;               // tile_dim1=K, tile_dim2=0
  g1[5] = 128;                      // tensor_dim0_stride=128 (lo32)
  g1[6] = 0;                        // stride0 hi | tensor_dim1_stride lo
  g1[7] = 0;
  const i32x4 z4 = {0, 0, 0, 0};
  const i32x8 z8 = {0, 0, 0, 0, 0, 0, 0, 0};
  __builtin_amdgcn_tensor_load_to_lds(g0, g1, z4, z4, z8, 0);
}
#else
#define HAVE_TDM 0
#endif

// Stage W (KR x 128 f32) into sB; call from all threads, ends synchronized.
template <int KR>
static __device__ __forceinline__ void stage_weight(float* sB, const float* W) {
#if HAVE_TDM
  if (threadIdx.x < 32) {  // wave 0 issues the DMA and waits on TENSORcnt
    tdm_load_Kx128(W, (unsigned)(uintptr_t)sB, KR);
    __builtin_amdgcn_s_wait_tensorcnt(0);
  }
  __syncthreads();
#else
  for (int i = threadIdx.x; i < KR * DIM / 4; i += 256)
    ((float4*)sB)[i] = ((const float4*)W)[i];
  __syncthreads();
#endif
}

// Wave computes D(16 x 128) = A(16 x K) * B(K x 128); acc[t] = cols [16t,16t+16).
// Lane mapping (ISA 7.12.2): lr = lane&15, lh = lane>>4.
//   A elem (lane, v) = A[lr][k0 + lh*2 + v]           -> one float2 load
//   B elem (lane, v) = B[k0 + lh*2 + v][t*16 + lr]    -> ds_load_2addr pair
//   C elem (lane, g, t) = D[lh*8 + g][t*16 + lr]
template <int K, class AF, class BF>
static __device__ __forceinline__ void wave_gemm(AF loadA, BF loadB, v8f acc[8]) {
#pragma unroll 4
  for (int k0 = 0; k0 < K; k0 += 4) {
    v2f a = loadA(k0);
#pragma unroll
    for (int t = 0; t < 8; ++t) {
      v2f b = loadB(k0, t);
      acc[t] = wmma4(a, b, acc[t]);
    }
  }
}

// ---------------- Kernel 1: hn = h @ W_node (N x 128) ----------------------
__global__ void __launch_bounds__(256) k_node_gemm(const float* __restrict__ h,
                                                   const float* __restrict__ Wn,
                                                   float* __restrict__ hn) {
  __shared__ float sB[DIM * DIM];
  stage_weight<DIM>(sB, Wn);

  const int lane = threadIdx.x & 31, wave = threadIdx.x >> 5;
  const int lr = lane & 15, lh = lane >> 4;
  const int tile = blockIdx.x * 8 + wave;
  if (tile * 16 >= N_NODES) return;
  const int rowBase = tile * 16;
  const float* arow = h + (size_t)(rowBase + lr) * DIM;

  v8f acc[8] = {};
  wave_gemm<DIM>(
      [&](int k0) -> v2f { return *(const v2f*)(arow + k0 + lh * 2); },
      [&](int k0, int t) -> v2f {
        const float* p = sB + (k0 + lh * 2) * DIM + t * 16 + lr;
        v2f b; b.x = p[0]; b.y = p[DIM]; return b;
      },
      acc);

#pragma unroll
  for (int g = 0; g < 8; ++g) {
    const int r = rowBase + lh * 8 + g;
#pragma unroll
    for (int t = 0; t < 8; ++t) hn[(size_t)r * DIM + t * 16 + lr] = acc[t][g];
  }
}

// ------- Kernel 2: ow = edge_w @ W_eedge; w1 = (lrelu(ow)@W_esf)*ow --------
// Writes ow to workspace and w1 directly into sat[:, 256:384] (d_out slab).
__global__ void __launch_bounds__(256) k_edge_gemm(const float* __restrict__ ew,
                                                   const float* __restrict__ We,
                                                   const float* __restrict__ Wesf,
                                                   float* __restrict__ ow,
                                                   float* __restrict__ sat) {
  __shared__ float sB[DIM * DIM];
  stage_weight<DIM>(sB, We);

  const int lane = threadIdx.x & 31, wave = threadIdx.x >> 5;
  const int lr = lane & 15, lh = lane >> 4;
  const int tile = blockIdx.x * 8 + wave;
  if (tile * 16 >= N_EDGES) return;
  const int rowBase = tile * 16;
  const float* arow = ew + (size_t)(rowBase + lr) * DIM;

  v8f acc[8] = {};
  wave_gemm<DIM>(
      [&](int k0) -> v2f { return *(const v2f*)(arow + k0 + lh * 2); },
      [&](int k0, int t) -> v2f {
        const float* p = sB + (k0 + lh * 2) * DIM + t * 16 + lr;
        v2f b; b.x = p[0]; b.y = p[DIM]; return b;
      },
      acc);

  // row scalar: s_r = sum_c lrelu(ow[r,c]) * W_esf[c]
  float wv[8];
#pragma unroll
  for (int t = 0; t < 8; ++t) wv[t] = Wesf[t * 16 + lr];
  float p[8];
#pragma unroll
  for (int g = 0; g < 8; ++g) {
    float s = 0.f;
#pragma unroll
    for (int t = 0; t < 8; ++t) s += lrelu(acc[t][g]) * wv[t];
    p[g] = s;
  }
#pragma unroll
  for (int m = 8; m >= 1; m >>= 1)
#pragma unroll
    for (int g = 0; g < 8; ++g) p[g] += __shfl_xor(p[g], m, 32);

#pragma unroll
  for (int g = 0; g < 8; ++g) {
    const int r = rowBase + lh * 8 + g;
#pragma unroll
    for (int t = 0; t < 8; ++t) {
      const int c = t * 16 + lr;
      const float o = acc[t][g];
      ow[(size_t)r * DIM + c] = o;
      sat[(size_t)r * 384 + 256 + c] = p[g] * o;  // w1
    }
  }
}

// -------- Kernel 3: sat[:,0:128]=hn[src], sat[:,128:256]=hn[dst] -----------
__global__ void __launch_bounds__(256) k_gather(const float* __restrict__ hn,
                                                const int* __restrict__ src,
                                                const int* __restrict__ dst,
                                                float* __restrict__ sat) {
  const int e = blockIdx.x * 8 + (threadIdx.x >> 5);
  const int c4 = (threadIdx.x & 31) * 4;
  if (e >= N_EDGES) return;
  float4 a = *(const float4*)(hn + (size_t)src[e] * DIM + c4);
  float4 b = *(const float4*)(hn + (size_t)dst[e] * DIM + c4);
  *(float4*)(sat + (size_t)e * 384 + c4) = a;
  *(float4*)(sat + (size_t)e * 384 + 128 + c4) = b;
}

// -- Kernel 4: inw = lrelu(sat@W_fuse+b); a = inw@W_attn; w2 = a*w1; s = a --
__global__ void __launch_bounds__(256) k_fuse(const float* __restrict__ sat,
                                              const float* __restrict__ Wf,
                                              const float* __restrict__ bf,
                                              const float* __restrict__ Wa,
                                              float* __restrict__ sbuf,
                                              float* __restrict__ w2) {
  const int lane = threadIdx.x & 31, wave = threadIdx.x >> 5;
  const int lr = lane & 15, lh = lane >> 4;
  const int tile = blockIdx.x * 8 + wave;
  if (tile * 16 >= N_EDGES) return;
  const int rowBase = tile * 16;
  const float* arow = sat + (size_t)(rowBase + lr) * 384;

  v8f acc[8] = {};
  wave_gemm<384>(
      [&](int k0) -> v2f { return *(const v2f*)(arow + k0 + lh * 2); },
      [&](int k0, int t) -> v2f {
        const float* p = Wf + (k0 + lh * 2) * DIM + t * 16 + lr;  // hot in WGP$/L2
        v2f b; b.x = p[0]; b.y = p[DIM]; return b;
      },
      acc);

  float bv[8], wv[8];
#pragma unroll
  for (int t = 0; t < 8; ++t) { bv[t] = bf[t * 16 + lr]; wv[t] = Wa[t * 16 + lr]; }
  float p[8];
#pragma unroll
  for (int g = 0; g < 8; ++g) {
    float s = 0.f;
#pragma unroll
    for (int t = 0; t < 8; ++t) s += lrelu(acc[t][g] + bv[t]) * wv[t];
    p[g] = s;
  }
#pragma unroll
  for (int m = 8; m >= 1; m >>= 1)
#pragma unroll
    for (int g = 0; g < 8; ++g) p[g] += __shfl_xor(p[g], m, 32);

#pragma unroll
  for (int g = 0; g < 8; ++g) {
    const int r = rowBase + lh * 8 + g;
    if (lr == 0) sbuf[r] = p[g];
#pragma unroll
    for (int t = 0; t < 8; ++t) {
      const int c = t * 16 + lr;
      w2[(size_t)r * DIM + c] = p[g] * sat[(size_t)r * 384 + 256 + c];
    }
  }
}

// ----------------- Kernel 5: edge softmax + scatter ------------------------
static __device__ __forceinline__ unsigned fkey(float x) {
  unsigned u = __float_as_uint(x);
  return (u & 0x80000000u) ? ~u : (u | 0x80000000u);
}
static __device__ __forceinline__ float unfkey(unsigned k) {
  unsigned u = (k & 0x80000000u) ? (k & 0x7fffffffu) : ~k;
  return __uint_as_float(u);
}

__global__ void k_init(unsigned* __restrict__ mkey, float* __restrict__ denom,
                       float* __restrict__ indeg, float* __restrict__ hagg) {
  const long i = (long)blockIdx.x * 256 + threadIdx.x;
  if (i < N_NODES) { mkey[i] = 0x007FFFFFu; denom[i] = 0.f; indeg[i] = 0.f; }
  if (i < (long)N_NODES * 256) hagg[i] = 0.f;
}

__global__ void k_smax(const float* __restrict__ s, const int* __restrict__ dst,
                       unsigned* __restrict__ mkey) {
  const int e = blockIdx.x * 256 + threadIdx.x;
  if (e >= N_EDGES) return;
  atomicMax(&mkey[dst[e]], fkey(s[e]));
}

__global__ void k_sexp(const float* __restrict__ s, const int* __restrict__ dst,
                       const unsigned* __restrict__ mkey, float* __restrict__ ev,
                       float* __restrict__ denom, float* __restrict__ indeg) {
  const int e = blockIdx.x * 256 + threadIdx.x;
  if (e >= N_EDGES) return;
  const int d = dst[e];
  const float v = __expf(s[e] - unfkey(mkey[d]));
  ev[e] = v;
  atomicAdd(&denom[d], v);
  atomicAdd(&indeg[d], 1.0f);
}

// h_agg[dst] += alpha * [hs | w2]  (256 cols per edge, 4 cols per thread)
__global__ void k_scatter(const float* __restrict__ sat, const float* __restrict__ w2,
                          const float* __restrict__ ev, const float* __restrict__ denom,
                          const int* __restrict__ dst, float* __restrict__ hagg) {
  const long idx = (long)blockIdx.x * 256 + threadIdx.x;
  if (idx >= (long)N_EDGES * 64) return;
  const int e = (int)(idx >> 6);
  const int c4 = ((int)idx & 63) << 2;
  const int d = dst[e];
  const float alpha = ev[e] / denom[d];
  float4 t;
  if (c4 < 128) t = *(const float4*)(sat + (size_t)e * 384 + c4);
  else          t = *(const float4*)(w2 + (size_t)e * DIM + (c4 - 128));
  float* p = hagg + (size_t)d * 256 + c4;
  atomicAdd(p + 0, alpha * t.x);
  atomicAdd(p + 1, alpha * t.y);
  atomicAdd(p + 2, alpha * t.z);
  atomicAdd(p + 3, alpha * t.w);
}

// ---- Kernel 6: h_out = indeg>0 ? [h_agg|hn]@W_conc + b_conc : hn ----------
__global__ void __launch_bounds__(256) k_conc(const float* __restrict__ hagg,
                                              const float* __restrict__ hn,
                                              const float* __restrict__ Wc,
                                              const float* __restrict__ bc,
                                              const float* __restrict__ indeg,
                                              float* __restrict__ hout) {
  const int lane = threadIdx.x & 31, wave = threadIdx.x >> 5;
  const int lr = lane & 15, lh = lane >> 4;
  const int tile = blockIdx.x * 8 + wave;
  if (tile * 16 >= N_NODES) return;
  const int rowBase = tile * 16;
  const float* ar0 = hagg + (size_t)(rowBase + lr) * 256;
  const float* ar1 = hn + (size_t)(rowBase + lr) * DIM;

  v8f acc[8] = {};
  wave_gemm<384>(
      [&](int k0) -> v2f {
        const int kk = k0 + lh * 2;
        return kk < 256 ? *(const v2f*)(ar0 + kk) : *(const v2f*)(ar1 + (kk - 256));
      },
      [&](int k0, int t) -> v2f {
        const float* p = Wc + (k0 + lh * 2) * DIM + t * 16 + lr;
        v2f b; b.x = p[0]; b.y = p[DIM]; return b;
      },
      acc);

  float bv[8];
#pragma unroll
  for (int t = 0; t < 8; ++t) bv[t] = bc[t * 16 + lr];
#pragma unroll
  for (int g = 0; g < 8; ++g) {
    const int r = rowBase + lh * 8 + g;
    const float deg = indeg[r];
#pragma unroll
    for (int t = 0; t < 8; ++t) {
      const int c = t * 16 + lr;
      const float v = acc[t][g] + bv[t];
      hout[(size_t)r * DIM + c] = (deg > 0.f) ? v : hn[(size_t)r * DIM + c];
    }
  }
}

// ---- Kernel 7: w_out = [h_out[src]|h_out[dst]|w_bn|ow] @ W_aggre ----------
__global__ void __launch_bounds__(256) k_aggre(const float* __restrict__ hout,
                                               const float* __restrict__ w2,
                                               const float* __restrict__ ow,
                                               const float* __restrict__ gamma,
                                               const float* __restrict__ beta,
                                               const float* __restrict__ Wagg,
                                               const int* __restrict__ src,
                                               const int* __restrict__ dst,
                                               float* __restrict__ wout) {
  const int lane = threadIdx.x & 31, wave = threadIdx.x >> 5;
  const int lr = lane & 15, lh = lane >> 4;
  const int tile = blockIdx.x * 8 + wave;
  if (tile * 16 >= N_EDGES) return;
  const int rowBase = tile * 16;
  const int e = rowBase + lr;
  const float* ps = hout + (size_t)src[e] * DIM;
  const float* pd = hout + (size_t)dst[e] * DIM;
  const float* pw = w2 + (size_t)e * DIM;
  const float* po = ow + (size_t)e * DIM;
  const float invs = rsqrtf(1.0f + 1e-5f);

  v8f acc[8] = {};
  wave_gemm<512>(
      [&](int k0) -> v2f {
        const int kk = k0 + lh * 2;  // segments at 0/128/256/384 (mult. of 4)
        v2f a;
        if (kk < 128) a = *(const v2f*)(ps + kk);
        else if (kk < 256) a = *(const v2f*)(pd + (kk - 128));
        else if (kk < 384) {
          const int c = kk - 256;
          v2f w = *(const v2f*)(pw + c);
          v2f g = *(const v2f*)(gamma + c);
          v2f b = *(const v2f*)(beta + c);
          a.x = w.x * invs * g.x + b.x;
          a.y = w.y * invs * g.y + b.y;
        } else a = *(const v2f*)(po + (kk - 384));
        return a;
      },
      [&](int k0, int t) -> v2f {
        const float* p = Wagg + (k0 + lh * 2) * DIM + t * 16 + lr;
        v2f b; b.x = p[0]; b.y = p[DIM]; return b;
      },
      acc);

#pragma unroll
  for (int g = 0; g < 8; ++g) {
    const int r = rowBase + lh * 8 + g;
#pragma unroll
    for (int t = 0; t < 8; ++t) wout[(size_t)r * DIM + t * 16 + lr] = acc[t][g];
  }
}

// ---------------------------------------------------------------------------
extern "C" void kernel_launch(void* const* d_in, const int* in_sizes, int n_in,
                              void* d_out, int out_size, void* d_ws, size_t ws_size,
                              hipStream_t stream) {
  const float* h       = (const float*)d_in[0];
  const float* edge_w  = (const float*)d_in[1];
  const float* W_node  = (const float*)d_in[2];
  const float* W_eedge = (const float*)d_in[3];
  const float* W_esf   = (const float*)d_in[4];
  const float* W_fuse  = (const float*)d_in[5];
  const float* b_fuse  = (const float*)d_in[6];
  const float* W_attn  = (const float*)d_in[7];
  const float* W_conc  = (const float*)d_in[8];
  const float* b_conc  = (const float*)d_in[9];
  const float* gamma   = (const float*)d_in[10];
  const float* beta    = (const float*)d_in[11];
  const float* W_aggre = (const float*)d_in[12];
  const int*   src     = (const int*)d_in[13];
  const int*   dst     = (const int*)d_in[14];
  (void)in_sizes; (void)n_in; (void)out_size; (void)ws_size;

  float* out  = (float*)d_out;
  float* hout = out;                               // N x 128
  float* wout = hout + (size_t)N_NODES * DIM;      // E x 128
  float* sat  = wout + (size_t)N_EDGES * DIM;      // E x 384

  float* ws = (float*)d_ws;                        // ~361 MB total
  float* hn    = ws; ws += (size_t)N_NODES * DIM;
  float* ow    = ws; ws += (size_t)N_EDGES * DIM;
  float* w2    = ws; ws += (size_t)N_EDGES * DIM;
  float* sbuf  = ws; ws += N_EDGES;
  float* ev    = ws; ws += N_EDGES;
  float* denom = ws; ws += N_NODES;
  float* indeg = ws; ws += N_NODES;
  unsigned* mkey = (unsigned*)ws; ws += N_NODES;
  float* hagg  = ws; ws += (size_t)N_NODES * 256;

  const int tilesN = N_NODES / 16, tilesE = N_EDGES / 16;
  dim3 blk(256);
  k_node_gemm<<<(tilesN + 7) / 8, blk, 0, stream>>>(h, W_node, hn);
  k_edge_gemm<<<(tilesE + 7) / 8, blk, 0, stream>>>(edge_w, W_eedge, W_esf, ow, sat);
  k_gather<<<N_EDGES / 8, blk, 0, stream>>>(hn, src, dst, sat);
  k_fuse<<<(tilesE + 7) / 8, blk, 0, stream>>>(sat, W_fuse, b_fuse, W_attn, sbuf, w2);
  k_init<<<N_NODES, blk, 0, stream>>>(mkey, denom, indeg, hagg);
  k_smax<<<(N_EDGES + 255) / 256, blk, 0, stream>>>(sbuf, dst, mkey);
  k_sexp<<<(N_EDGES + 255) / 256, blk, 0, stream>>>(sbuf, dst, mkey, ev, denom, indeg);
  k_scatter<<<(int)(((long)N_EDGES * 64 + 255) / 256), blk, 0, stream>>>(sat, w2, ev, denom, dst, hagg);
  k_conc<<<(tilesN + 7) / 8, blk, 0, stream>>>(hagg, hn, W_conc, b_conc, indeg, hout);
  k_aggre<<<(tilesE + 7) / 8, blk, 0, stream>>>(hout, w2, ow, gamma, beta, W_aggre, src, dst, wout);
}